// noflayer_65901978190181
// MI455X (gfx1250) — compile-verified
//
#include <hip/hip_runtime.h>

#define D 128

typedef __attribute__((ext_vector_type(2))) float v2f;
typedef __attribute__((ext_vector_type(8))) float v8f;

// ---- order-preserving float<->uint key for atomicMax-based segment max ----
__device__ __forceinline__ unsigned int fkey(float f) {
  unsigned int b = __float_as_uint(f);
  return (b & 0x80000000u) ? ~b : (b | 0x80000000u);
}
__device__ __forceinline__ float funkey(unsigned int u) {
  unsigned int b = (u & 0x80000000u) ? (u ^ 0x80000000u) : ~u;
  return __uint_as_float(b);
}
#define KEY_NEG_INF 0x007FFFFFu  // fkey(-inf)

// ---- init: zero accumulators, set row-max keys to -inf ----
__global__ void k_init(float* __restrict__ s, float* __restrict__ rowsum,
                       unsigned int* __restrict__ mkey,
                       float* __restrict__ feat_even, float* __restrict__ out,
                       int nNodes) {
  long long idx = (long long)blockIdx.x * blockDim.x + threadIdx.x;
  long long tot = (long long)nNodes * D;
  if (idx < tot) { feat_even[idx] = 0.0f; out[idx] = 0.0f; }
  if (idx < nNodes) { s[idx] = 0.0f; rowsum[idx] = 0.0f; mkey[idx] = KEY_NEG_INF; }
}

// ---- build B panel (column-major, 16 columns x 128 K):
// col 0 = W_att @ a[:D], col 1 = W_att @ a[D:], cols 2..15 = 0.
// bmat[n*D + k] = B[k][n].  8KB total, L2-resident, shared by all WMMA waves.
__global__ void k_wa(const float* __restrict__ W, const float* __restrict__ a,
                     float* __restrict__ bmat) {
  int k = threadIdx.x;
  const float* wr = W + k * D;
  float s1 = 0.0f, s2 = 0.0f;
  for (int j = 0; j < D; ++j) {
    float w = wr[j];
    s1 += w * a[j];
    s2 += w * a[D + j];
  }
  bmat[0 * D + k] = s1;
  bmat[1 * D + k] = s2;
#pragma unroll
  for (int n = 2; n < 16; ++n) bmat[n * D + k] = 0.0f;
}

// ---- f1/f2 via V_WMMA_F32_16X16X4_F32: one wave per 16-node tile ----
// A: 16x4 f32 tile of input rows. B: 4x16 panel (from bmat, branchless per-lane ptr).
// D(16x16): col 0 = f1 for the 16 nodes, col 1 = f2.
__global__ __launch_bounds__(128) void k_f1f2(const float* __restrict__ input,
                                              const float* __restrict__ bmat,
                                              float* __restrict__ f1,
                                              float* __restrict__ f2,
                                              int nTiles, int nNodes) {
  int wave = threadIdx.x >> 5;
  int lane = threadIdx.x & 31;
  int tile = blockIdx.x * 4 + wave;
  if (tile >= nTiles) return;          // wave-uniform: EXEC stays all-1s for WMMA

  int n  = lane & 15;                  // A: M index / B,C,D: N index
  int hi = lane >> 4;                  // selects K pair {0,1} vs {2,3} (A/B), M+8 (C/D)
  int node = tile * 16 + n;
  if (node >= nNodes) node = nNodes - 1;   // clamp tail loads (100000 % 16 == 0 anyway)
  const float* arow = input + (size_t)node * D;   // per-lane A row
  const float* bcol = bmat + (size_t)n * D;       // per-lane B column (branchless)

  v8f acc = {};
#pragma unroll 8
  for (int k0 = 0; k0 < D; k0 += 4) {
    int kk = k0 + 2 * hi;              // ISA layout: lanes 0-15 K={0,1}, 16-31 K={2,3}
    v2f A = *(const v2f*)(arow + kk);  // 8B-aligned (kk even)
    v2f B = *(const v2f*)(bcol + kk);
    acc = __builtin_amdgcn_wmma_f32_16x16x4_f32(
        /*neg_a=*/false, A, /*neg_b=*/false, B,
        /*c_mod=*/(short)0, acc, /*reuse_a=*/false, /*reuse_b=*/false);
  }

  // C/D layout: lane holds column (lane&15) for rows M = hi*8 + v, v=0..7
  int mbase = tile * 16 + hi * 8;
  if (n == 0) {
#pragma unroll
    for (int v = 0; v < 8; ++v)
      if (mbase + v < nNodes) f1[mbase + v] = acc[v];
  } else if (n == 1) {
#pragma unroll
    for (int v = 0; v < 8; ++v)
      if (mbase + v < nNodes) f2[mbase + v] = acc[v];
  }
}

// ---- edge pass 1: e = lrelu(f1[r] + f2[c]); segment max via atomicMax on keys ----
__global__ void k_edge1(const int* __restrict__ rows, const int* __restrict__ cols,
                        const float* __restrict__ f1, const float* __restrict__ f2,
                        float* __restrict__ e_ws, unsigned int* __restrict__ mkey,
                        int nEdges) {
  int i = blockIdx.x * blockDim.x + threadIdx.x;
  if (i >= nEdges) return;
  int r = rows[i], c = cols[i];
  float e = f1[r] + f2[c];
  e = e > 0.0f ? e : 0.2f * e;
  e_ws[i] = e;
  atomicMax(&mkey[r], fkey(e));
}

// ---- edge pass 2: ex = exp(e - m[r]); segment sum ----
__global__ void k_edge2(const int* __restrict__ rows,
                        const unsigned int* __restrict__ mkey,
                        float* __restrict__ e_ws, float* __restrict__ s,
                        int nEdges) {
  int i = blockIdx.x * blockDim.x + threadIdx.x;
  if (i >= nEdges) return;
  int r = rows[i];
  float ex = __expf(e_ws[i] - funkey(mkey[r]));
  e_ws[i] = ex;
  unsafeAtomicAdd(&s[r], ex);
}

// ---- edge pass 3 (heavy, fused): one wave32 per edge.
// gather g = input[c] (512B coalesced, L2-resident); scatter att*g -> feat_even[r],
// adj*g -> out (adj_mm accumulator), adj*0.5*att -> rowsum[r].
__global__ __launch_bounds__(256) void k_edge3(
    const int* __restrict__ rows, const int* __restrict__ cols,
    const float* __restrict__ adj, const float* __restrict__ ex_ws,
    const float* __restrict__ s, const float* __restrict__ input,
    float* __restrict__ feat_even, float* __restrict__ out_acc,
    float* __restrict__ rowsum, int nEdges) {
  int wave = (int)((blockIdx.x * blockDim.x + threadIdx.x) >> 5);
  int lane = threadIdx.x & 31;
  if (wave >= nEdges) return;
  int r = rows[wave];
  int c = cols[wave];
  float att = ex_ws[wave] / s[r];
  float av  = adj[wave];
  const float4* g4 = (const float4*)(input + (size_t)c * D);
  float4 g = g4[lane];
  float* fe = feat_even + (size_t)r * D + lane * 4;
  float* oa = out_acc   + (size_t)r * D + lane * 4;
  unsafeAtomicAdd(&fe[0], att * g.x);
  unsafeAtomicAdd(&fe[1], att * g.y);
  unsafeAtomicAdd(&fe[2], att * g.z);
  unsafeAtomicAdd(&fe[3], att * g.w);
  unsafeAtomicAdd(&oa[0], av * g.x);
  unsafeAtomicAdd(&oa[1], av * g.y);
  unsafeAtomicAdd(&oa[2], av * g.z);
  unsafeAtomicAdd(&oa[3], av * g.w);
  if (lane == 0) unsafeAtomicAdd(&rowsum[r], 0.5f * av * att);
}

// ---- epilogue: out = 0.15*adj_mm + (0.35 - 0.15*rowsum)*feat_even + 0.5*h0 ----
__global__ void k_final(const float* __restrict__ feat_even,
                        const float* __restrict__ rowsum,
                        const float* __restrict__ h0, float* __restrict__ out,
                        int nNodes) {
  long long idx = (long long)blockIdx.x * blockDim.x + threadIdx.x;
  long long tot = (long long)nNodes * D;
  if (idx >= tot) return;
  int node = (int)(idx >> 7);
  float fe = feat_even[idx];
  float am = out[idx];                    // adj_mm was accumulated in-place in d_out
  float rs = rowsum[node];
  out[idx] = 0.15f * am + (0.35f - 0.15f * rs) * fe + 0.5f * h0[idx];
}

extern "C" void kernel_launch(void* const* d_in, const int* in_sizes, int n_in,
                              void* d_out, int out_size, void* d_ws, size_t ws_size,
                              hipStream_t stream) {
  const float* input = (const float*)d_in[0];
  const float* h0    = (const float*)d_in[1];
  const int*   rows  = (const int*)d_in[2];
  const int*   cols  = (const int*)d_in[3];
  const float* adj   = (const float*)d_in[4];
  const float* W     = (const float*)d_in[5];
  const float* a     = (const float*)d_in[6];
  // d_in[7..9] = lamda, alpha, l : unused by the reference math

  int nNodes = in_sizes[0] / D;
  int nEdges = in_sizes[2];
  float* out = (float*)d_out;

  // ws layout (floats): bmat[16*D] | f1[N] | f2[N] | mkey[N] | s[N] | rowsum[N] | e[E] | feat_even[N*D]
  float* ws        = (float*)d_ws;
  float* bmat      = ws;
  float* f1        = bmat + 16 * D;
  float* f2        = f1 + nNodes;
  unsigned int* mk = (unsigned int*)(f2 + nNodes);
  float* s         = (float*)(mk + nNodes);
  float* rowsum    = s + nNodes;
  float* e_ws      = rowsum + nNodes;
  float* feat_even = e_ws + nEdges;

  long long tot = (long long)nNodes * D;
  int tb = 256;
  int nodeBlocks = (int)((tot + tb - 1) / tb);
  int edgeBlocks = (nEdges + tb - 1) / tb;

  k_init<<<nodeBlocks, tb, 0, stream>>>(s, rowsum, mk, feat_even, out, nNodes);
  k_wa<<<1, D, 0, stream>>>(W, a, bmat);

  int nTiles = (nNodes + 15) / 16;
  k_f1f2<<<(nTiles + 3) / 4, 128, 0, stream>>>(input, bmat, f1, f2, nTiles, nNodes);

  k_edge1<<<edgeBlocks, tb, 0, stream>>>(rows, cols, f1, f2, e_ws, mk, nEdges);
  k_edge2<<<edgeBlocks, tb, 0, stream>>>(rows, mk, e_ws, s, nEdges);

  int waveBlocks = (nEdges + 7) / 8;  // one wave32 per edge, 8 waves per block
  k_edge3<<<waveBlocks, tb, 0, stream>>>(rows, cols, adj, e_ws, s, input,
                                         feat_even, out, rowsum, nEdges);

  k_final<<<nodeBlocks, tb, 0, stream>>>(feat_even, rowsum, h0, out, nNodes);
}